// JDEBBoxPostProcess_58377195487337
// MI455X (gfx1250) — compile-verified
//
#include <hip/hip_runtime.h>
#include <hip/hip_bf16.h>

#define KBOX   4096
#define NWORDS 128          // KBOX / 32
#define NMS_T  0.6f

typedef __attribute__((ext_vector_type(2))) float v2f;
typedef __attribute__((ext_vector_type(8))) float v8f;

// ---------------------------------------------------------------------------
// Kernel A: gather 4096 boxes (SoA) from the 4M x 5 tensor
// ---------------------------------------------------------------------------
__global__ void gather_kernel(const float* __restrict__ yolo,
                              const int*   __restrict__ bidx,
                              float* __restrict__ gx1, float* __restrict__ gy1,
                              float* __restrict__ gx2, float* __restrict__ gy2,
                              float* __restrict__ gsc) {
  int k = blockIdx.x * blockDim.x + threadIdx.x;
  if (k >= KBOX) return;
  long long b = (long long)bidx[k] * 5;
  gx1[k] = yolo[b + 0];
  gy1[k] = yolo[b + 1];
  gx2[k] = yolo[b + 2];
  gy2[k] = yolo[b + 3];
  gsc[k] = yolo[b + 4];
}

// ---------------------------------------------------------------------------
// Kernel B: single-workgroup bitonic sort of 4096 (score, idx) keys in LDS.
// Descending score, ascending index tiebreak => matches stable argsort(-s).
// ---------------------------------------------------------------------------
__global__ __launch_bounds__(1024)
void sort_kernel(const float* __restrict__ gx1, const float* __restrict__ gy1,
                 const float* __restrict__ gx2, const float* __restrict__ gy2,
                 const float* __restrict__ gsc,
                 float* __restrict__ sx1, float* __restrict__ sy1,
                 float* __restrict__ sx2, float* __restrict__ sy2,
                 float* __restrict__ ss,  float* __restrict__ areas,
                 int*   __restrict__ order) {
  __shared__ float sk[KBOX];
  __shared__ int   si[KBOX];
  const int tid = threadIdx.x;

  for (int i = tid; i < KBOX; i += 1024) { sk[i] = gsc[i]; si[i] = i; }
  __syncthreads();

  for (int k = 2; k <= KBOX; k <<= 1) {
    for (int j = k >> 1; j > 0; j >>= 1) {
      for (int i = tid; i < KBOX; i += 1024) {
        int ixj = i ^ j;
        if (ixj > i) {
          float a = sk[i], b = sk[ixj];
          int   ai = si[i], bi = si[ixj];
          // after(a,b): a belongs after b in final (descending) order
          bool a_after_b = (a < b) || (a == b && ai > bi);
          bool b_after_a = (b < a) || (b == a && bi > ai);
          bool up = ((i & k) == 0);
          if (up ? a_after_b : b_after_a) {
            sk[i] = b; sk[ixj] = a;
            si[i] = bi; si[ixj] = ai;
          }
        }
      }
      __syncthreads();
    }
  }

  for (int i = tid; i < KBOX; i += 1024) {
    int kk = si[i];
    order[i] = kk;
    float sc = sk[i];  ss[i] = sc;
    float x1 = gx1[kk], y1 = gy1[kk], x2 = gx2[kk], y2 = gy2[kk];
    sx1[i] = x1; sy1[i] = y1; sx2[i] = x2; sy2[i] = y2;
    areas[i] = (x2 - x1 + 1.0f) * (y2 - y1 + 1.0f);
  }
}

// ---------------------------------------------------------------------------
// Kernel C: suppression bitmask. One wave computes a 16-row x 32-col tile.
// Union base (areas[i] + areas[j]) via V_WMMA_F32_16X16X4_F32 (rank-2 outer
// sum: A = [area_row | 1], B = [1 ; area_col]). WMMA D layout: lane -> N,
// vgpr r -> M = r + 8*(lane/16), which lines up with wave32 ballots: each
// ballot packs 16 bits of row r (lanes 0-15) and 16 bits of row r+8.
// ---------------------------------------------------------------------------
__global__ __launch_bounds__(256)
void mask_kernel(const float* __restrict__ sx1, const float* __restrict__ sy1,
                 const float* __restrict__ sx2, const float* __restrict__ sy2,
                 const float* __restrict__ areas,
                 unsigned* __restrict__ mask) {
  __shared__ float lrx1[16], lry1[16], lrx2[16], lry2[16], lrar[16];
  __shared__ float lcx1[256], lcy1[256], lcx2[256], lcy2[256], lcar[256];

  const int tid = threadIdx.x;
  const int rb  = (blockIdx.x >> 4) * 16;    // 16 rows per block
  const int cb  = (blockIdx.x & 15) * 256;   // 256 cols per block (8 waves x 32)

  lcx1[tid] = sx1[cb + tid];
  lcy1[tid] = sy1[cb + tid];
  lcx2[tid] = sx2[cb + tid];
  lcy2[tid] = sy2[cb + tid];
  lcar[tid] = areas[cb + tid];
  if (tid < 16) {
    lrx1[tid] = sx1[rb + tid];
    lry1[tid] = sy1[rb + tid];
    lrx2[tid] = sx2[rb + tid];
    lry2[tid] = sy2[rb + tid];
    lrar[tid] = areas[rb + tid];
  }
  __syncthreads();

  const int wave  = tid >> 5;
  const int lane  = tid & 31;
  const int half  = lane >> 4;       // 0: lanes 0-15, 1: lanes 16-31
  const int ln    = lane & 15;
  const int word  = (blockIdx.x & 15) * 8 + wave;  // global 32-col word index
  const int wbase = wave * 32;                     // local col base in LDS

  // A 16x4 f32 layout: lanes 0-15 hold K=0 (v0), K=1 (v1); lanes 16-31 K=2,3.
  // B 4x16 f32 layout: lanes 0-15 hold K=0 (v0), K=1 (v1); lanes 16-31 K=2,3.
  const float hsel = half ? 0.0f : 1.0f;
  v2f A;  A.x  = hsel * lrar[ln];            A.y  = hsel;
  v2f B0; B0.x = hsel;  B0.y = hsel * lcar[wbase + ln];
  v2f B1; B1.x = hsel;  B1.y = hsel * lcar[wbase + 16 + ln];
  v8f zero = {};

  // d[r] at lane = areas[row rb + r + 8*half] + areas[col]
  v8f d0 = __builtin_amdgcn_wmma_f32_16x16x4_f32(false, A, false, B0,
                                                 (short)0, zero, false, false);
  v8f d1 = __builtin_amdgcn_wmma_f32_16x16x4_f32(false, A, false, B1,
                                                 (short)0, zero, false, false);

  const float c0x1 = lcx1[wbase + ln],      c0y1 = lcy1[wbase + ln];
  const float c0x2 = lcx2[wbase + ln],      c0y2 = lcy2[wbase + ln];
  const float c1x1 = lcx1[wbase + 16 + ln], c1y1 = lcy1[wbase + 16 + ln];
  const float c1x2 = lcx2[wbase + 16 + ln], c1y2 = lcy2[wbase + 16 + ln];

  unsigned myword = 0u;
#pragma unroll
  for (int r = 0; r < 8; ++r) {
    const int m = r + 8 * half;
    const float rx1 = lrx1[m], ry1 = lry1[m], rx2 = lrx2[m], ry2 = lry2[m];

    float w0 = fmaxf(0.0f, fminf(rx2, c0x2) - fmaxf(rx1, c0x1) + 1.0f);
    float h0 = fmaxf(0.0f, fminf(ry2, c0y2) - fmaxf(ry1, c0y1) + 1.0f);
    float in0 = w0 * h0;
    float w1 = fmaxf(0.0f, fminf(rx2, c1x2) - fmaxf(rx1, c1x1) + 1.0f);
    float h1 = fmaxf(0.0f, fminf(ry2, c1y2) - fmaxf(ry1, c1y1) + 1.0f);
    float in1 = w1 * h1;

    // inter/union >= T  <=>  inter >= T * (area_i + area_j - inter); union > 0
    bool p0 = in0 >= NMS_T * (d0[r] - in0);
    bool p1 = in1 >= NMS_T * (d1[r] - in1);

    unsigned b0 = __builtin_amdgcn_ballot_w32(p0);
    unsigned b1 = __builtin_amdgcn_ballot_w32(p1);
    unsigned wA = (b0 & 0xFFFFu) | (b1 << 16);          // row r
    unsigned wB = (b0 >> 16) | (b1 & 0xFFFF0000u);      // row r + 8
    if (lane == r)     myword = wA;
    if (lane == r + 8) myword = wB;
  }

  if (lane < 16)
    mask[(size_t)(rb + lane) * NWORDS + word] = myword;
}

// ---------------------------------------------------------------------------
// Kernel D: single-wave greedy suppression scan. Suppression vector lives in
// 4 VGPRs per lane (128 dwords); keep-bit broadcast via ds_bpermute. Only
// kept rows pull their (L2-resident) mask row.
// ---------------------------------------------------------------------------
__global__ __launch_bounds__(32)
void reduce_kernel(const unsigned* __restrict__ mask,
                   unsigned* __restrict__ keepw,
                   float* __restrict__ bbox_num_out) {
  const int lane = threadIdx.x;
  unsigned rem0 = 0u, rem1 = 0u, rem2 = 0u, rem3 = 0u;
  unsigned kp0 = 0u, kp1 = 0u, kp2 = 0u, kp3 = 0u;

  for (int i = 0; i < KBOX; ++i) {
    const int wi = i >> 5, bit = i & 31;
    const int owner = wi & 31;        // lane holding word wi
    const int rsel  = wi >> 5;        // which register (uniform)
    unsigned rv = (rsel == 0) ? rem0 : (rsel == 1) ? rem1
                : (rsel == 2) ? rem2 : rem3;
    unsigned w = (unsigned)__builtin_amdgcn_ds_bpermute(owner << 2, (int)rv);
    bool kept = ((w >> bit) & 1u) == 0u;      // uniform across the wave
    if (kept) {
      const unsigned* row = mask + (size_t)i * NWORDS;
      rem0 |= row[lane];
      rem1 |= row[lane + 32];
      rem2 |= row[lane + 64];
      rem3 |= row[lane + 96];
      if (lane == owner) {
        if (rsel == 0) kp0 |= 1u << bit;
        else if (rsel == 1) kp1 |= 1u << bit;
        else if (rsel == 2) kp2 |= 1u << bit;
        else kp3 |= 1u << bit;
      }
    }
  }

  keepw[lane]      = kp0;
  keepw[lane + 32] = kp1;
  keepw[lane + 64] = kp2;
  keepw[lane + 96] = kp3;

  int cnt = __popc(kp0) + __popc(kp1) + __popc(kp2) + __popc(kp3);
#pragma unroll
  for (int off = 16; off > 0; off >>= 1)
    cnt += __builtin_amdgcn_ds_bpermute((lane ^ off) << 2, cnt);
  if (lane == 0) *bbox_num_out = (float)cnt;
}

// ---------------------------------------------------------------------------
// Kernel E: emit bbox_pred (K x 6), nms_keep_idx (as float; ints cast)
// ---------------------------------------------------------------------------
__global__ void finalize_kernel(const unsigned* __restrict__ keepw,
                                const float* __restrict__ ss,
                                const float* __restrict__ sx1,
                                const float* __restrict__ sy1,
                                const float* __restrict__ sx2,
                                const float* __restrict__ sy2,
                                const int*   __restrict__ order,
                                float* __restrict__ out) {
  int i = blockIdx.x * blockDim.x + threadIdx.x;
  if (i >= KBOX) return;
  bool kept = (keepw[i >> 5] >> (i & 31)) & 1u;
  float m = kept ? 1.0f : 0.0f;
  float* bp = out + (size_t)i * 6;
  bp[0] = 0.0f;          // label
  bp[1] = ss[i] * m;
  bp[2] = sx1[i] * m;
  bp[3] = sy1[i] * m;
  bp[4] = sx2[i] * m;
  bp[5] = sy2[i] * m;
  out[KBOX * 6 + 1 + i] = (float)(kept ? order[i] : -1);
}

// ---------------------------------------------------------------------------
extern "C" void kernel_launch(void* const* d_in, const int* in_sizes, int n_in,
                              void* d_out, int out_size, void* d_ws, size_t ws_size,
                              hipStream_t stream) {
  const float* yolo = (const float*)d_in[0];   // (4000000, 5) f32
  const int*   bidx = (const int*)d_in[1];     // (4096,) i32
  float* out = (float*)d_out;                  // 4096*6 + 1 + 4096 floats

  // workspace layout (float units)
  float* f = (float*)d_ws;
  float* gx1 = f;           float* gy1 = f + 4096;
  float* gx2 = f + 8192;    float* gy2 = f + 12288;
  float* gsc = f + 16384;
  float* sx1 = f + 20480;   float* sy1 = f + 24576;
  float* sx2 = f + 28672;   float* sy2 = f + 32768;
  float* ss  = f + 36864;   float* areas = f + 40960;
  int*      order = (int*)(f + 45056);
  unsigned* keepw = (unsigned*)(f + 49152);
  unsigned* mask  = (unsigned*)(f + 49280);    // 4096*128 dwords = 2 MB

  gather_kernel<<<16, 256, 0, stream>>>(yolo, bidx, gx1, gy1, gx2, gy2, gsc);
  sort_kernel<<<1, 1024, 0, stream>>>(gx1, gy1, gx2, gy2, gsc,
                                      sx1, sy1, sx2, sy2, ss, areas, order);
  mask_kernel<<<4096, 256, 0, stream>>>(sx1, sy1, sx2, sy2, areas, mask);
  reduce_kernel<<<1, 32, 0, stream>>>(mask, keepw, out + KBOX * 6);
  finalize_kernel<<<16, 256, 0, stream>>>(keepw, ss, sx1, sy1, sx2, sy2,
                                          order, out);
}